// CenterNet_44092134261241
// MI455X (gfx1250) — compile-verified
//
#include <hip/hip_runtime.h>

// CenterNet decode for MI455X (gfx1250, wave32).
// Memory-bound op (~176 MB traffic -> ~7.6 us floor @ 23.3 TB/s). No matmul
// => no WMMA; the CDNA5 feature that applies is the Tensor Data Mover:
// each 128x128 heat plane is DMA'd into LDS with tensor_load_to_lds and
// synchronized via s_wait_tensorcnt (TENSORcnt path).

#define Bn 32
#define Cn 80
#define Hd 128
#define Wd 128
#define HW (Hd * Wd)       // 16384
#define Kn 100
#define THREADS 256
#define PER_T (HW / THREADS) // 64 pixels per thread
#define POOL (Cn * Kn)       // 8000

typedef unsigned int v4u __attribute__((ext_vector_type(4)));
typedef int          v8i __attribute__((ext_vector_type(8)));
typedef int          v4i __attribute__((ext_vector_type(4)));

// ---- wave32 u64 max reduction via 32-bit shuffles (no u64 overload risk) ----
__device__ __forceinline__ unsigned long long shfl_xor_u64(unsigned long long v, int m) {
    int lo = (int)(unsigned)v;
    int hi = (int)(unsigned)(v >> 32);
    lo = __shfl_xor(lo, m, 32);
    hi = __shfl_xor(hi, m, 32);
    return ((unsigned long long)(unsigned)hi << 32) | (unsigned)lo;
}

__device__ __forceinline__ unsigned long long wave_max_u64(unsigned long long v) {
#pragma unroll
    for (int off = 16; off > 0; off >>= 1) {
        unsigned long long o = shfl_xor_u64(v, off);
        v = (o > v) ? o : v;
    }
    return v;
}

// Map float to order-preserving unsigned key (finite inputs).
__device__ __forceinline__ unsigned ord_key(float f) {
    unsigned u = __float_as_uint(f);
    return (u & 0x80000000u) ? ~u : (u | 0x80000000u);
}

// ============================================================================
// Stage 1: one block per (b,c) plane.
//   TDM-load plane to LDS -> NMS peak keys in registers -> 100 rounds of
//   block argmax -> scores_pc (sigmoid bits) + inds_pc (flat spatial idx).
// ============================================================================
__global__ __launch_bounds__(THREADS) void centernet_stage1(
    const float* __restrict__ heat,
    unsigned* __restrict__ out_sc,   // [B, C*K] sigmoid bits (rank order)
    unsigned* __restrict__ out_ix)   // [B, C*K] flat spatial idx
{
    __shared__ __align__(16) float plane[HW];   // 64 KB (WGP has 320 KB LDS)

    const int pid = blockIdx.x;                 // b*C + c
    const float* __restrict__ src = heat + (size_t)pid * HW;
    const int t = threadIdx.x;

    // ---- Stage plane into LDS with the Tensor Data Mover ----
#if __has_builtin(__builtin_amdgcn_tensor_load_to_lds)
    if (t == 0) {
        unsigned long long ga = (unsigned long long)(uintptr_t)src;      // 57-bit
        unsigned lds_off = (unsigned)(uintptr_t)(void*)plane;            // low 32 = LDS byte offset

        v4u g0;
        g0.x = 1u;                                   // count=1, user desc
        g0.y = lds_off;                              // lds_addr
        g0.z = (unsigned)(ga & 0xFFFFFFFFu);         // global_addr[31:0]
        g0.w = (unsigned)((ga >> 32) & 0x01FFFFFFu)  // global_addr[56:32]
               | (2u << 30);                         // type = 2 ("image")

        v8i g1;
        g1[0] = 0x00020000;                 // wg_mask=0, data_size=2 (4B)
        g1[1] = (int)((unsigned)Wd << 16);  // tensor_dim0[15:0] in bits 63:48
        g1[2] = (int)((unsigned)Hd << 16);  // tensor_dim1[15:0] in bits 95:80
        g1[3] = (int)((unsigned)Wd << 16);  // tile_dim0 in bits 127:112
        g1[4] = Hd;                         // tile_dim1 (bits 143:128); tile_dim2=0
        g1[5] = Wd;                         // tensor_dim0_stride[31:0]
        g1[6] = 0;                          // stride hi / dim1_stride lo
        g1[7] = 0;
        v4i z4 = {0, 0, 0, 0};              // 2D tensor: groups 2/3 unused
        v8i z8 = {0, 0, 0, 0, 0, 0, 0, 0};
        __builtin_amdgcn_tensor_load_to_lds(g0, g1, z4, z4, z8, 0);
#if __has_builtin(__builtin_amdgcn_s_wait_tensorcnt)
        __builtin_amdgcn_s_wait_tensorcnt(0);
#else
        asm volatile("s_wait_tensorcnt 0" ::: "memory");
#endif
    }
#else
    for (int i = t; i < HW / 4; i += THREADS)
        ((float4*)plane)[i] = ((const float4*)src)[i];
#endif
    __syncthreads();

    // ---- 3x3 NMS on raw heat (sigmoid is monotonic => same peak set) ----
    // Thread t owns pixels idx = t + 256*j (bank-conflict-free LDS access).
    unsigned sk[PER_T];
#pragma unroll
    for (int j = 0; j < PER_T; ++j) {
        const int idx = t + (j << 8);
        const int y = idx >> 7, x = idx & (Wd - 1);
        const float c = plane[idx];
        float nmax = -3.402823466e38f;
        if (y > 0) {
            if (x > 0)      nmax = fmaxf(nmax, plane[idx - Wd - 1]);
                            nmax = fmaxf(nmax, plane[idx - Wd]);
            if (x < Wd - 1) nmax = fmaxf(nmax, plane[idx - Wd + 1]);
        }
        if (x > 0)          nmax = fmaxf(nmax, plane[idx - 1]);
        if (x < Wd - 1)     nmax = fmaxf(nmax, plane[idx + 1]);
        if (y < Hd - 1) {
            if (x > 0)      nmax = fmaxf(nmax, plane[idx + Wd - 1]);
                            nmax = fmaxf(nmax, plane[idx + Wd]);
            if (x < Wd - 1) nmax = fmaxf(nmax, plane[idx + Wd + 1]);
        }
        sk[j] = (c >= nmax) ? ord_key(c) : 0u;   // suppressed -> score 0
    }
    __syncthreads();   // plane no longer needed; reuse its LDS as scratch

    unsigned long long* red = reinterpret_cast<unsigned long long*>(plane);
    unsigned m0 = 0xFFFFFFFFu, m1 = 0xFFFFFFFFu;   // "not yet taken" bits
    const int wave = t >> 5, lane = t & 31;
    unsigned* __restrict__ outs = out_sc + (size_t)pid * Kn;
    unsigned* __restrict__ outi = out_ix + (size_t)pid * Kn;

    for (int k = 0; k < Kn; ++k) {
        unsigned long long best = 0;
#pragma unroll
        for (int j = 0; j < PER_T; ++j) {
            const unsigned avail = (j < 32) ? (m0 >> j) : (m1 >> (j - 32));
            if (avail & 1u) {
                const int idx = t + (j << 8);
                const unsigned long long key =
                    ((unsigned long long)sk[j] << 32) | (unsigned)(HW - 1 - idx);
                if (key > best) best = key;
            }
        }
        best = wave_max_u64(best);
        if (lane == 0) red[wave] = best;
        __syncthreads();
        if (t == 0) {
            unsigned long long w = red[0];
#pragma unroll
            for (int i = 1; i < THREADS / 32; ++i) w = (red[i] > w) ? red[i] : w;
            red[8] = w;
            const unsigned widx = (unsigned)(HW - 1) - (unsigned)(w & 0xFFFFFFFFu);
            const unsigned su = (unsigned)(w >> 32);
            float score = 0.0f;
            if (su) {   // invert ord_key, sigmoid only for winners
                const unsigned rb = (su & 0x80000000u) ? (su & 0x7FFFFFFFu) : ~su;
                const float raw = __uint_as_float(rb);
                score = 1.0f / (1.0f + __expf(-raw));
            }
            outs[k] = __float_as_uint(score);
            outi[k] = widx;
        }
        __syncthreads();
        const unsigned widx = (unsigned)(HW - 1) - (unsigned)(red[8] & 0xFFFFFFFFu);
        if ((int)(widx & (THREADS - 1)) == t) {
            const int j = (int)(widx >> 8);
            if (j < 32) m0 &= ~(1u << j);
            else        m1 &= ~(1u << (j - 32));
        }
    }
}

// ============================================================================
// Stage 2: one block per batch. Global top-100 over the 8000 pooled entries
// (key embeds pooled index => same tie-break as reference), gather wh/reg,
// write dets [B, K, 6].
// ============================================================================
__global__ __launch_bounds__(THREADS) void centernet_stage2(
    const unsigned* __restrict__ sc,    // [B, 8000]
    const unsigned* __restrict__ ix,    // [B, 8000]
    const float* __restrict__ wh,       // [B, 2, H, W]
    const float* __restrict__ reg,      // [B, 2, H, W]
    const float* __restrict__ thrs,     // [C]
    float* __restrict__ out)            // [B, K, 6]
{
    __shared__ __align__(16) unsigned long long red[9];
    const int b = blockIdx.x;
    const int t = threadIdx.x;
    const unsigned* __restrict__ sb = sc + (size_t)b * POOL;

    unsigned skv[32];
    unsigned avail = 0;
#pragma unroll
    for (int s = 0; s < 32; ++s) {
        const int j = t + (s << 8);
        if (j < POOL) { skv[s] = sb[j]; avail |= (1u << s); }
        else            skv[s] = 0u;
    }

    for (int k = 0; k < Kn; ++k) {
        unsigned long long best = 0;
#pragma unroll
        for (int s = 0; s < 32; ++s) {
            if ((avail >> s) & 1u) {
                const int j = t + (s << 8);
                const unsigned long long key =
                    ((unsigned long long)skv[s] << 13) | (unsigned)(8191 - j);
                if (key > best) best = key;
            }
        }
        best = wave_max_u64(best);
        if ((t & 31) == 0) red[t >> 5] = best;
        __syncthreads();
        if (t == 0) {
            unsigned long long w = red[0];
#pragma unroll
            for (int i = 1; i < THREADS / 32; ++i) w = (red[i] > w) ? red[i] : w;
            red[8] = w;

            const int j = 8191 - (int)(w & 0x1FFFu);     // pooled index
            const float score = __uint_as_float((unsigned)(w >> 13));
            const int cls = j / Kn;
            const unsigned sp = ix[(size_t)b * POOL + j]; // flat spatial idx
            const float ysf = (float)(sp >> 7);
            const float xsf = (float)(sp & (Wd - 1));
            const size_t base = (size_t)b * 2 * HW + sp;
            const float w0 = wh[base],  w1 = wh[base + HW];
            const float r0 = reg[base], r1 = reg[base + HW];
            const float xc = xsf + r0,  yc = ysf + r1;
            const float hx = 0.5f * w0, hy = 0.5f * w1;
            const float th = thrs[cls];
            const float clsf = (score < th) ? -1.0f : (float)cls;
            float* o = out + ((size_t)b * Kn + k) * 6;
            o[0] = clsf; o[1] = score;
            o[2] = xc - hx; o[3] = yc - hy;
            o[4] = xc + hx; o[5] = yc + hy;
        }
        __syncthreads();
        const int j = 8191 - (int)(red[8] & 0x1FFFu);
        if ((j & (THREADS - 1)) == t) avail &= ~(1u << (j >> 8));
    }
}

extern "C" void kernel_launch(void* const* d_in, const int* in_sizes, int n_in,
                              void* d_out, int out_size, void* d_ws, size_t ws_size,
                              hipStream_t stream) {
    const float* heat = (const float*)d_in[0];   // [32,80,128,128]
    const float* wh   = (const float*)d_in[1];   // [32,2,128,128]
    const float* reg  = (const float*)d_in[2];   // [32,2,128,128]
    const float* thrs = (const float*)d_in[3];   // [80]
    // d_in[4] = K (always 100 per reference K_DEFAULT; compile-time constant)

    unsigned* sc = (unsigned*)d_ws;                       // [32][8000] score bits
    unsigned* ix = sc + (size_t)Bn * POOL;                // [32][8000] spatial idx
    (void)in_sizes; (void)n_in; (void)out_size; (void)ws_size;

    centernet_stage1<<<Bn * Cn, THREADS, 0, stream>>>(heat, sc, ix);
    centernet_stage2<<<Bn, THREADS, 0, stream>>>(sc, ix, wh, reg, thrs, (float*)d_out);
}